// BPNP_fast_34445637714639
// MI455X (gfx1250) — compile-verified
//
#include <hip/hip_runtime.h>
#include <math.h>

typedef float v2f __attribute__((ext_vector_type(2)));
typedef float v8f __attribute__((ext_vector_type(8)));

#define WPB   4        // waves per block (wave32)
#define NPTS  128
#define ROWS  256      // 2*NPTS residual rows
#define COLS  8        // 7 used ([J(6) | r]) + 1 zero pad column
#define NITER 8
#define DAMP  1e-4f

__global__ __launch_bounds__(WPB * 32)
void bpnp_lm_wmma_kernel(const float* __restrict__ pts2d,
                         const float* __restrict__ pts3d,
                         const float* __restrict__ Kmat,
                         const float* __restrict__ init_pose,
                         float* __restrict__ out,
                         int batch)
{
    // Column-major Jacobian staging per wave: JL[col][row], col 7 = zeros.
    __shared__ float JL[WPB][COLS * ROWS];
    __shared__ float Hb[WPB][7 * 16];

    const int lane  = threadIdx.x & 31;
    const int wslot = threadIdx.x >> 5;
    const int inst0 = blockIdx.x * WPB + wslot;
    const bool active = (inst0 < batch);
    const int inst = active ? inst0 : (batch - 1);   // clamp: keep barriers uniform

    // Intrinsics (uniform)
    const float fx  = Kmat[0];
    const float cxk = Kmat[2];
    const float fy  = Kmat[4];
    const float cyk = Kmat[5];

    // Pose replicated across lanes
    const float* pp = init_pose + (size_t)inst * 6;
    float p6[6];
#pragma unroll
    for (int i = 0; i < 6; ++i) p6[i] = pp[i];

    // Each lane owns 4 points: p = lane + 32*q (coalesced)
    float Xp[4], Yp[4], Zp[4], Uo[4], Vo[4];
#pragma unroll
    for (int q = 0; q < 4; ++q) {
        int p = lane + 32 * q;
        const float* p3 = pts3d + ((size_t)inst * NPTS + p) * 3;
        Xp[q] = p3[0]; Yp[q] = p3[1]; Zp[q] = p3[2];
        const float* p2 = pts2d + ((size_t)inst * NPTS + p) * 2;
        Uo[q] = p2[0]; Vo[q] = p2[1];
    }

    const int m    = lane & 15;
    const int colc = (m < 7) ? m : 7;                 // clamp into zero-pad column
    const int kh   = (lane >> 4) << 1;                // 0 for lanes 0-15, 2 for 16-31
    const float* cb = &JL[wslot][colc * ROWS];

    for (int it = 0; it < NITER; ++it) {
        // ---------- Rodrigues + analytic dR/dr_j (redundant per lane) ----------
        float r_[3] = { p6[0], p6[1], p6[2] };
        float th2 = r_[0]*r_[0] + r_[1]*r_[1] + r_[2]*r_[2] + 1e-12f;
        float th  = sqrtf(th2);
        float ith = 1.0f / th;
        float k_[3] = { r_[0]*ith, r_[1]*ith, r_[2]*ith };
        float s = sinf(th), c = cosf(th), omc = 1.0f - c;

        float Km[3][3] = { { 0.f,  -k_[2],  k_[1] },
                           { k_[2],  0.f,  -k_[0] },
                           { -k_[1], k_[0],  0.f } };
        float K2m[3][3];
#pragma unroll
        for (int i = 0; i < 3; ++i)
#pragma unroll
            for (int j = 0; j < 3; ++j)
                K2m[i][j] = k_[i] * k_[j] - ((i == j) ? 1.0f : 0.0f);

        float R[3][3];
#pragma unroll
        for (int i = 0; i < 3; ++i)
#pragma unroll
            for (int j = 0; j < 3; ++j)
                R[i][j] = ((i == j) ? 1.0f : 0.0f) + s * Km[i][j] + omc * K2m[i][j];

        float dR[3][3][3];
#pragma unroll
        for (int j = 0; j < 3; ++j) {
            float dth = r_[j] * ith;                       // dtheta/dr_j
            float dk[3];
#pragma unroll
            for (int a = 0; a < 3; ++a)
                dk[a] = ((a == j) ? ith : 0.0f) - k_[a] * dth * ith;
            float dK[3][3] = { { 0.f,  -dk[2],  dk[1] },
                               { dk[2],  0.f,  -dk[0] },
                               { -dk[1], dk[0],  0.f } };
#pragma unroll
            for (int i = 0; i < 3; ++i)
#pragma unroll
                for (int l = 0; l < 3; ++l) {
                    float dK2 = 0.0f;
#pragma unroll
                    for (int a = 0; a < 3; ++a)
                        dK2 += dK[i][a] * Km[a][l] + Km[i][a] * dK[a][l];
                    dR[j][i][l] = c * dth * Km[i][l] + s * dK[i][l]
                                + s * dth * K2m[i][l] + omc * dK2;
                }
        }

        // ---------- Per-point residual + Jacobian rows -> LDS ----------
#pragma unroll
        for (int q = 0; q < 4; ++q) {
            float x = Xp[q], y = Yp[q], z3 = Zp[q];
            float cam0 = R[0][0]*x + R[0][1]*y + R[0][2]*z3 + p6[3];
            float cam1 = R[1][0]*x + R[1][1]*y + R[1][2]*z3 + p6[4];
            float cam2 = R[2][0]*x + R[2][1]*y + R[2][2]*z3 + p6[5];
            float invz  = 1.0f / cam2;
            float xz = cam0 * invz, yz = cam1 * invz;
            float ru = fx * xz + cxk - Uo[q];
            float rv = fy * yz + cyk - Vo[q];

            float Ju[7], Jv[7];
#pragma unroll
            for (int j = 0; j < 3; ++j) {
                float d0 = dR[j][0][0]*x + dR[j][0][1]*y + dR[j][0][2]*z3;
                float d1 = dR[j][1][0]*x + dR[j][1][1]*y + dR[j][1][2]*z3;
                float d2 = dR[j][2][0]*x + dR[j][2][1]*y + dR[j][2][2]*z3;
                Ju[j] = fx * invz * (d0 - xz * d2);
                Jv[j] = fy * invz * (d1 - yz * d2);
            }
            Ju[3] = fx * invz;           Jv[3] = 0.0f;
            Ju[4] = 0.0f;                Jv[4] = fy * invz;
            Ju[5] = -fx * xz * invz;     Jv[5] = -fy * yz * invz;
            Ju[6] = ru;                  Jv[6] = rv;

            int p = lane + 32 * q;
            int ur = 2 * p, vr = 2 * p + 1;
#pragma unroll
            for (int col = 0; col < 7; ++col) {
                JL[wslot][col * ROWS + ur] = Ju[col];
                JL[wslot][col * ROWS + vr] = Jv[col];
            }
            JL[wslot][7 * ROWS + ur] = 0.0f;   // zero pad column
            JL[wslot][7 * ROWS + vr] = 0.0f;
        }
        __syncthreads();

        // ---------- Gram matrix M = [J|r]^T [J|r] via V_WMMA_F32_16X16X4_F32 ----------
        // A (16x4) and B (4x16) fragments hold identical data for A^T A: reuse one load.
        v8f acc = {0.f,0.f,0.f,0.f,0.f,0.f,0.f,0.f};
#pragma unroll 4
        for (int ch = 0; ch < 64; ++ch) {
            int r0 = (ch << 2) + kh;
            v2f f;
            f.x = cb[r0];
            f.y = cb[r0 + 1];
            acc = __builtin_amdgcn_wmma_f32_16x16x4_f32(
                false, f, false, f, (short)0, acc, false, false);
        }
        __syncthreads();

        // ---------- Spill H|g rows 0..6 (lanes 0..15 hold N=0..15 of VGPR i = row i) ----------
        if (lane < 16) {
#pragma unroll
            for (int i = 0; i < 7; ++i)
                Hb[wslot][i * 16 + lane] = acc[i];
        }
        __syncthreads();

        // ---------- Redundant 6x6 damped solve on every lane ----------
        float A_[6][7];
#pragma unroll
        for (int i = 0; i < 6; ++i) {
#pragma unroll
            for (int j = 0; j < 6; ++j)
                A_[i][j] = Hb[wslot][i * 16 + j] + ((i == j) ? DAMP : 0.0f);
            A_[i][6] = Hb[wslot][i * 16 + 6];   // g = J^T r
        }
#pragma unroll
        for (int kk = 0; kk < 6; ++kk) {
            float inv = 1.0f / A_[kk][kk];
#pragma unroll
            for (int i = 0; i < 6; ++i) {
                if (i > kk) {
                    float fkt = A_[i][kk] * inv;
#pragma unroll
                    for (int j = 0; j < 7; ++j)
                        if (j >= kk) A_[i][j] -= fkt * A_[kk][j];
                }
            }
        }
        float xs[6];
#pragma unroll
        for (int ii = 5; ii >= 0; --ii) {
            float v = A_[ii][6];
#pragma unroll
            for (int j = 0; j < 6; ++j)
                if (j > ii) v -= A_[ii][j] * xs[j];
            xs[ii] = v / A_[ii][ii];
        }
#pragma unroll
        for (int i = 0; i < 6; ++i) p6[i] -= xs[i];

        __syncthreads();   // JL/Hb reuse fence before next iteration
    }

    if (active && lane == 0) {
        float* o = out + (size_t)inst * 6;
#pragma unroll
        for (int i = 0; i < 6; ++i) o[i] = p6[i];
    }
}

extern "C" void kernel_launch(void* const* d_in, const int* in_sizes, int n_in,
                              void* d_out, int out_size, void* d_ws, size_t ws_size,
                              hipStream_t stream) {
    const float* pts2d     = (const float*)d_in[0];   // [B,128,2]
    const float* pts3d     = (const float*)d_in[1];   // [B,128,3]
    const float* Kmat      = (const float*)d_in[2];   // [3,3]
    const float* init_pose = (const float*)d_in[3];   // [B,6]
    float* out = (float*)d_out;                       // [B,6]

    int batch = in_sizes[3] / 6;
    int blocks = (batch + WPB - 1) / WPB;
    bpnp_lm_wmma_kernel<<<blocks, WPB * 32, 0, stream>>>(
        pts2d, pts3d, Kmat, init_pose, out, batch);
}